// GraphSAGE_86225763434551
// MI455X (gfx1250) — compile-verified
//
#include <hip/hip_runtime.h>

typedef float v2f __attribute__((ext_vector_type(2)));
typedef float v8f __attribute__((ext_vector_type(8)));

#define N_NODES 100000
#define N_EDGES 1600000
#define DIM 64
#define TILES_PER_WAVE 10   // 6250 row tiles = 625 blocks * 10

// ---------------- helpers ----------------

__global__ __launch_bounds__(256) void fill_f32(float* __restrict__ p, float v, int n) {
    int i = blockIdx.x * blockDim.x + threadIdx.x;
    if (i < n) p[i] = v;
}

__global__ __launch_bounds__(256) void deg_count(const long long* __restrict__ dst,
                                                 float* __restrict__ deg) {
    int e = blockIdx.x * blockDim.x + threadIdx.x;
    if (e < N_EDGES) atomicAdd(&deg[(int)dst[e]], 1.0f);
}

__global__ __launch_bounds__(256) void inv_deg(const float* __restrict__ deg,
                                               float* __restrict__ invdeg) {
    int i = blockIdx.x * blockDim.x + threadIdx.x;
    if (i < N_NODES) invdeg[i] = 1.0f / fmaxf(deg[i], 1.0f);
}

// One 16-lane group per edge; each lane moves 4 consecutive columns.
// 25,600,000 threads exactly == 100,000 blocks of 256 -> no bounds check.
__global__ __launch_bounds__(256) void scatter_add(const float* __restrict__ h,
                                                   const long long* __restrict__ src,
                                                   const long long* __restrict__ dst,
                                                   float* __restrict__ agg) {
    int gid = blockIdx.x * blockDim.x + threadIdx.x;
    int e   = gid >> 4;
    int sub = gid & 15;
    int s = (int)src[e];
    int d = (int)dst[e];
    const float4 v = *(const float4*)(h + (long)s * DIM + sub * 4);
    float* a = agg + (long)d * DIM + sub * 4;
    atomicAdd(a + 0, v.x);
    atomicAdd(a + 1, v.y);
    atomicAdd(a + 2, v.z);
    atomicAdd(a + 3, v.w);
}

// ---------------- fused SAGE layer GEMM via f32 WMMA ----------------
// out = act( h @ Wself + (agg * invdeg_row) @ Wnei + bias )
// One wave per 16-wide output column block; wave keeps the B (weight)
// fragments for all 16 K-steps of both matrices register-resident
// (64 VGPRs) and sweeps TILES_PER_WAVE row tiles, 32 chained
// v_wmma_f32_16x16x4_f32 per tile.
template <int RELU>
__global__ __launch_bounds__(128) void sage_gemm(const float* __restrict__ h,
                                                 const float* __restrict__ agg,
                                                 const float* __restrict__ invdeg,
                                                 const float* __restrict__ wself,
                                                 const float* __restrict__ wnei,
                                                 const float* __restrict__ bias,
                                                 float* __restrict__ out) {
    int nb   = threadIdx.x >> 5;  // wave in block = column block 0..3
    int lane = threadIdx.x & 31;
    int half = lane >> 4;         // 0 or 1
    int lr   = lane & 15;
    int col  = nb * 16 + lr;      // B/C column (N) for this lane

    // Register-resident B fragments: B[k][col] pairs per ISA 4x16 layout
    // (lanes 16..31 hold K+2).
    v2f bs[16], bn[16];
#pragma unroll
    for (int ks = 0; ks < 16; ++ks) {
        int k = ks * 4 + half * 2;
        bs[ks].x = wself[k * DIM + col];
        bs[ks].y = wself[(k + 1) * DIM + col];
        bn[ks].x = wnei[k * DIM + col];
        bn[ks].y = wnei[(k + 1) * DIM + col];
    }
    float bb = bias[col];

#pragma unroll 1
    for (int t = 0; t < TILES_PER_WAVE; ++t) {
        int row_tile = blockIdx.x * TILES_PER_WAVE + t;  // 0..6249
        int row = row_tile * 16 + lr;                    // A-matrix row (M)
        float invd = invdeg[row];

        v8f c = {};
#pragma unroll
        for (int ks = 0; ks < 16; ++ks) {
            int k = ks * 4 + half * 2;

            // self term: A = h[row][k..k+1]
            v2f a0 = *(const v2f*)(h + (long)row * DIM + k);
            c = __builtin_amdgcn_wmma_f32_16x16x4_f32(false, a0, false, bs[ks],
                                                      (short)0, c, false, false);

            // neighbor term: A = agg[row][k..k+1] * invdeg[row]
            v2f a1 = *(const v2f*)(agg + (long)row * DIM + k);
            a1.x *= invd;
            a1.y *= invd;
            c = __builtin_amdgcn_wmma_f32_16x16x4_f32(false, a1, false, bn[ks],
                                                      (short)0, c, false, false);
        }

#pragma unroll
        for (int r = 0; r < 8; ++r) {
            // C/D layout: VGPR r = row (r + half*8), lane lr = column
            float v = c[r] + bb;
            if (RELU) v = fmaxf(v, 0.0f);
            out[(long)(row_tile * 16 + half * 8 + r) * DIM + col] = v;
        }
    }
}

// ---------------- driver ----------------

extern "C" void kernel_launch(void* const* d_in, const int* in_sizes, int n_in,
                              void* d_out, int out_size, void* d_ws, size_t ws_size,
                              hipStream_t stream) {
    const float*     x   = (const float*)d_in[0];
    const long long* ei  = (const long long*)d_in[1];  // int64 edge_index [2, E]
    const float*     ws1 = (const float*)d_in[2];
    const float*     wn1 = (const float*)d_in[3];
    const float*     b1  = (const float*)d_in[4];
    const float*     ws2 = (const float*)d_in[5];
    const float*     wn2 = (const float*)d_in[6];
    const float*     b2  = (const float*)d_in[7];
    const float*     ws3 = (const float*)d_in[8];
    const float*     wn3 = (const float*)d_in[9];
    const float*     b3  = (const float*)d_in[10];
    float*           out = (float*)d_out;

    const long long* src = ei;
    const long long* dst = ei + N_EDGES;

    char* w = (char*)d_ws;
    float* deg    = (float*)w;  w += (size_t)N_NODES * sizeof(float);
    float* invdeg = (float*)w;  w += (size_t)N_NODES * sizeof(float);
    float* agg    = (float*)w;  w += (size_t)N_NODES * DIM * sizeof(float);
    float* hA     = (float*)w;  w += (size_t)N_NODES * DIM * sizeof(float);
    float* hB     = (float*)w;

    const int NF = N_NODES * DIM;               // 6.4M floats
    const int FB = (NF + 255) / 256;            // fill blocks
    const int SB = (N_EDGES * 16) / 256;        // 100000 scatter blocks (exact)
    const int GB = (N_NODES / 16) / TILES_PER_WAVE;  // 625 gemm blocks of 4 waves

    // degree / inverse-degree (shared by all layers)
    fill_f32<<<(N_NODES + 255) / 256, 256, 0, stream>>>(deg, 0.0f, N_NODES);
    deg_count<<<(N_EDGES + 255) / 256, 256, 0, stream>>>(dst, deg);
    inv_deg<<<(N_NODES + 255) / 256, 256, 0, stream>>>(deg, invdeg);

    // layer 1: x -> hA
    fill_f32<<<FB, 256, 0, stream>>>(agg, 0.0f, NF);
    scatter_add<<<SB, 256, 0, stream>>>(x, src, dst, agg);
    sage_gemm<1><<<GB, 128, 0, stream>>>(x, agg, invdeg, ws1, wn1, b1, hA);

    // layer 2: hA -> hB
    fill_f32<<<FB, 256, 0, stream>>>(agg, 0.0f, NF);
    scatter_add<<<SB, 256, 0, stream>>>(hA, src, dst, agg);
    sage_gemm<1><<<GB, 128, 0, stream>>>(hA, agg, invdeg, ws2, wn2, b2, hB);

    // layer 3: hB -> out (no relu)
    fill_f32<<<FB, 256, 0, stream>>>(agg, 0.0f, NF);
    scatter_add<<<SB, 256, 0, stream>>>(hB, src, dst, agg);
    sage_gemm<0><<<GB, 128, 0, stream>>>(hB, agg, invdeg, ws3, wn3, b3, out);
}